// ChatSARAN_MLP_3513283248644
// MI455X (gfx1250) — compile-verified
//
#include <hip/hip_runtime.h>
#include <math.h>

// ---------------------------------------------------------------------------
// Fused last-token transformer block for MI455X (gfx1250, wave32).
//
// Key algebraic reduction: only x[:, -1, :] reaches the output, so
//   scores[b,s] = x[b,s] . u[b],  u[b] = Wk @ (Wq^T @ x_last[b])
//   attn_out[b] = (sum_s p[b,s] x[b,s]) @ Wv
// which removes the two big [8192x1024]x[1024x1024] GEMMs entirely.
// Remaining work is bandwidth-bound (~340 MB @ 23.3 TB/s ~ 15 us), so we
// stay in f32. MLP GEMVs use V_WMMA_F32_16X16X4_F32 (M padded 4->16).
// ---------------------------------------------------------------------------

#define Bq 4
#define Tq 2048
#define Dq 1024
#define Vq 50257
#define N8 8192
#define N4 4096

typedef float v2f __attribute__((ext_vector_type(2)));
typedef float v8f __attribute__((ext_vector_type(8)));

__device__ __forceinline__ float wave_sum(float v) {
#pragma unroll
  for (int off = 16; off > 0; off >>= 1) v += __shfl_xor(v, off, 32);
  return v;
}

// x_last[b,d] = tok_emb[idx[b,T-1],d] + pos_emb[T-1,d]
__global__ void k_embed_last(const int* __restrict__ idx, const float* __restrict__ te,
                             const float* __restrict__ pe, float* __restrict__ xl) {
  int b = blockIdx.y;
  int d = blockIdx.x * blockDim.x + threadIdx.x;
  int tok = idx[b * Tq + (Tq - 1)];
  xl[b * Dq + d] = te[(size_t)tok * Dq + d] + pe[(size_t)(Tq - 1) * Dq + d];
}

// y[b,n] = sum_k x[b,k] * W[k,n] (+ resid[b,n]); K = N = Dq. x row staged in LDS.
__global__ void k_gemv_col(const float* __restrict__ x, const float* __restrict__ W,
                           const float* __restrict__ resid, float* __restrict__ y) {
  __shared__ __align__(16) float xs[Dq];
  int b = blockIdx.y;
  int n = blockIdx.x * blockDim.x + threadIdx.x;
  for (int i = threadIdx.x; i < Dq; i += blockDim.x) xs[i] = x[b * Dq + i];
  __syncthreads();
  float acc = 0.f;
#pragma unroll 8
  for (int k = 0; k < Dq; ++k) acc = fmaf(xs[k], W[(size_t)k * Dq + n], acc);
  if (resid) acc += resid[b * Dq + n];
  y[b * Dq + n] = acc;
}

// u[b,d] = Wk[d,:] . q[b,:]   (one wave per row d, all 4 batches at once)
__global__ void k_gemv_row4(const float* __restrict__ W, const float* __restrict__ qv,
                            float* __restrict__ u) {
  int wave = (blockIdx.x * blockDim.x + threadIdx.x) >> 5;
  int lane = threadIdx.x & 31;
  if (wave >= Dq) return;
  const float* row = W + (size_t)wave * Dq;
  float a0 = 0.f, a1 = 0.f, a2 = 0.f, a3 = 0.f;
  for (int j = lane; j < Dq; j += 32) {
    float wv = row[j];
    a0 = fmaf(wv, qv[j], a0);
    a1 = fmaf(wv, qv[Dq + j], a1);
    a2 = fmaf(wv, qv[2 * Dq + j], a2);
    a3 = fmaf(wv, qv[3 * Dq + j], a3);
  }
  a0 = wave_sum(a0); a1 = wave_sum(a1); a2 = wave_sum(a2); a3 = wave_sum(a3);
  if (lane == 0) {
    u[wave] = a0; u[Dq + wave] = a1; u[2 * Dq + wave] = a2; u[3 * Dq + wave] = a3;
  }
}

// sc[b,s] = scale * (tok_emb[idx[b,s]] + pos_emb[s]) . u[b]   (one wave per (b,s))
__global__ void k_scores(const int* __restrict__ idx, const float* __restrict__ te,
                         const float* __restrict__ pe, const float* __restrict__ u,
                         float* __restrict__ sc) {
  int gw = (blockIdx.x * blockDim.x + threadIdx.x) >> 5;
  int lane = threadIdx.x & 31;
  int b = gw >> 11;           // / Tq
  int s = gw & (Tq - 1);
  const float* trow = te + (size_t)idx[b * Tq + s] * Dq;
  const float* prow = pe + (size_t)s * Dq;
  const float* ub = u + b * Dq;
  float acc = 0.f;
  for (int j = lane; j < Dq; j += 32) acc = fmaf(trow[j] + prow[j], ub[j], acc);
  acc = wave_sum(acc);
  if (lane == 0) sc[b * Tq + s] = acc * 0.03125f;  // 1/sqrt(1024)
}

// In-place stable softmax over each row of sc[B,T]; one block per b.
__global__ void k_softmax(float* __restrict__ sc) {
  __shared__ float red[256];
  float* row = sc + blockIdx.x * Tq;
  float vals[Tq / 256];
  float m = -INFINITY;
#pragma unroll
  for (int i = 0; i < Tq / 256; ++i) {
    vals[i] = row[threadIdx.x + 256 * i];
    m = fmaxf(m, vals[i]);
  }
  red[threadIdx.x] = m; __syncthreads();
  for (int off = 128; off > 0; off >>= 1) {
    if (threadIdx.x < off) red[threadIdx.x] = fmaxf(red[threadIdx.x], red[threadIdx.x + off]);
    __syncthreads();
  }
  m = red[0]; __syncthreads();
  float sum = 0.f;
#pragma unroll
  for (int i = 0; i < Tq / 256; ++i) { vals[i] = expf(vals[i] - m); sum += vals[i]; }
  red[threadIdx.x] = sum; __syncthreads();
  for (int off = 128; off > 0; off >>= 1) {
    if (threadIdx.x < off) red[threadIdx.x] += red[threadIdx.x + off];
    __syncthreads();
  }
  float inv = 1.f / red[0];
#pragma unroll
  for (int i = 0; i < Tq / 256; ++i) row[threadIdx.x + 256 * i] = vals[i] * inv;
}

// w[b,d] = sum_s p[b,s] * (tok_emb[idx[b,s],d] + pos_emb[s,d])
__global__ void k_wsum(const int* __restrict__ idx, const float* __restrict__ te,
                       const float* __restrict__ pe, const float* __restrict__ p,
                       float* __restrict__ w) {
  int b = blockIdx.y;
  int d = blockIdx.x * blockDim.x + threadIdx.x;
  float acc = 0.f;
  for (int s = 0; s < Tq; ++s) {
    float pv = p[b * Tq + s];
    int tok = idx[b * Tq + s];
    acc = fmaf(pv, te[(size_t)tok * Dq + d] + pe[(size_t)s * Dq + d], acc);
  }
  w[b * Dq + d] = acc;
}

// y[r,n0+l] = scaleOut*(sum_k x[r,k]W[k,n] + bias[n]) + resid[r,n]
// M padded 4->16 for V_WMMA_F32_16X16X4_F32; one wave per 16-wide N tile.
// A 16x4 layout (ISA 7.12.2): VGPR0 = {K0 | K2}, VGPR1 = {K1 | K3} across lane halves.
// B 4x16 mirrored; C: VGPR r -> row r (lanes 0-15) / row 8+r (lanes 16-31).
__global__ void k_gemv_wmma(const float* __restrict__ x, const float* __restrict__ W,
                            const float* __restrict__ bias, const float* __restrict__ resid,
                            float scaleOut, int K, int N, float* __restrict__ y) {
  int wave = (blockIdx.x * blockDim.x + threadIdx.x) >> 5;
  int lane = threadIdx.x & 31;
  int n0 = wave * 16;
  if (n0 >= N) return;                 // wave-uniform; grids sized exactly anyway
  int half = lane >> 4;
  int l16 = lane & 15;
  float amask = (l16 < 4) ? 1.f : 0.f; // rows 4..15 of padded A are zero
  const float* xr = x + (size_t)(l16 & 3) * K;
  v8f c = {};
  for (int k = 0; k < K; k += 4) {
    int ka = k + 2 * half;
    v2f a;
    a.x = amask * xr[ka];
    a.y = amask * xr[ka + 1];
    const float* wr = W + (size_t)ka * N + n0 + l16;
    v2f bm;
    bm.x = wr[0];
    bm.y = wr[(size_t)N];
    c = __builtin_amdgcn_wmma_f32_16x16x4_f32(false, a, false, bm, (short)0, c,
                                              false, false);
  }
  if (half == 0) {
#pragma unroll
    for (int r = 0; r < 4; ++r) {
      float v = c[r] + (bias ? bias[n0 + l16] : 0.f);
      v *= scaleOut;
      if (resid) v += resid[(size_t)r * N + n0 + l16];
      y[(size_t)r * N + n0 + l16] = v;
    }
  }
}

// act[b,j] = h[b,j] * gelu_exact(h[b,j+4096])
__global__ void k_geglu(const float* __restrict__ h, float* __restrict__ act) {
  int i = blockIdx.x * blockDim.x + threadIdx.x;  // [0, 4*4096)
  int b = i >> 12;
  int j = i & (N4 - 1);
  float a = h[(size_t)b * N8 + j];
  float g = h[(size_t)b * N8 + N4 + j];
  float gl = 0.5f * g * (1.f + erff(g * 0.70710678118654752f));
  act[i] = a * gl;
}

// out[b,v] = x_final[b,:] . tok_emb[v,:]  -- 206 MB stream, one wave per v,
// all 4 batches share the row load; x_final staged in LDS, B128 loads.
__global__ void k_logits(const float* __restrict__ xf, const float* __restrict__ te,
                         float* __restrict__ out) {
  __shared__ __align__(16) float xs[4 * Dq];
  for (int i = threadIdx.x; i < 4 * Dq; i += blockDim.x) xs[i] = xf[i];
  __syncthreads();
  int lane = threadIdx.x & 31;
  int v = blockIdx.x * (blockDim.x >> 5) + (threadIdx.x >> 5);
  if (v >= Vq) return;
  const float4* r4 = (const float4*)(te + (size_t)v * Dq);
  const float4* xs4 = (const float4*)xs;
  float a0 = 0.f, a1 = 0.f, a2 = 0.f, a3 = 0.f;
  for (int j = lane; j < Dq / 4; j += 32) {
    float4 t = r4[j];
    float4 x0 = xs4[j], x1 = xs4[256 + j], x2 = xs4[512 + j], x3 = xs4[768 + j];
    a0 = fmaf(t.x, x0.x, fmaf(t.y, x0.y, fmaf(t.z, x0.z, fmaf(t.w, x0.w, a0))));
    a1 = fmaf(t.x, x1.x, fmaf(t.y, x1.y, fmaf(t.z, x1.z, fmaf(t.w, x1.w, a1))));
    a2 = fmaf(t.x, x2.x, fmaf(t.y, x2.y, fmaf(t.z, x2.z, fmaf(t.w, x2.w, a2))));
    a3 = fmaf(t.x, x3.x, fmaf(t.y, x3.y, fmaf(t.z, x3.z, fmaf(t.w, x3.w, a3))));
  }
  a0 = wave_sum(a0); a1 = wave_sum(a1); a2 = wave_sum(a2); a3 = wave_sum(a3);
  if (lane == 0) {
    out[v] = a0;
    out[Vq + v] = a1;
    out[2 * Vq + v] = a2;
    out[3 * Vq + v] = a3;
  }
}

extern "C" void kernel_launch(void* const* d_in, const int* in_sizes, int n_in,
                              void* d_out, int out_size, void* d_ws, size_t ws_size,
                              hipStream_t stream) {
  (void)in_sizes; (void)n_in; (void)out_size; (void)ws_size;
  const int*   idx = (const int*)d_in[0];
  const float* te  = (const float*)d_in[1];
  const float* pe  = (const float*)d_in[2];
  const float* Wq  = (const float*)d_in[3];
  const float* Wk  = (const float*)d_in[4];
  const float* Wv  = (const float*)d_in[5];
  const float* W1  = (const float*)d_in[6];
  const float* b1  = (const float*)d_in[7];
  const float* W2  = (const float*)d_in[8];
  const float* b2  = (const float*)d_in[9];
  float* out = (float*)d_out;
  float* ws  = (float*)d_ws;

  // Workspace layout (floats): total 320 KB
  float* xl   = ws;          // [4,1024] x_last
  float* qv   = ws + 4096;   // [4,1024] q at last token
  float* u    = ws + 8192;   // [4,1024] u = Wk @ q
  float* xnew = ws + 12288;  // [4,1024] x_last + attn_out
  float* xfin = ws + 16384;  // [4,1024] final hidden
  float* sc   = ws + 20480;  // [4,2048] scores -> probs (in place)
  float* wacc = ws + 28672;  // [4,1024] attn-weighted sum of x rows
  float* h    = ws + 32768;  // [4,8192] MLP pre-activation
  float* act  = ws + 65536;  // [4,4096] GEGLU output

  k_embed_last<<<dim3(Dq / 256, Bq), 256, 0, stream>>>(idx, te, pe, xl);
  k_gemv_col<<<dim3(Dq / 256, Bq), 256, 0, stream>>>(xl, Wq, nullptr, qv);
  k_gemv_row4<<<(Dq * 32) / 256, 256, 0, stream>>>(Wk, qv, u);
  k_scores<<<(Bq * Tq * 32) / 256, 256, 0, stream>>>(idx, te, pe, u, sc);
  k_softmax<<<Bq, 256, 0, stream>>>(sc);
  k_wsum<<<dim3(Dq / 256, Bq), 256, 0, stream>>>(idx, te, pe, sc, wacc);
  k_gemv_col<<<dim3(Dq / 256, Bq), 256, 0, stream>>>(wacc, Wv, xl, xnew);
  // h = xnew @ W1 + b1     (512 N-tiles -> 64 blocks of 8 waves)
  k_gemv_wmma<<<(N8 / 16) / 8, 256, 0, stream>>>(xnew, W1, b1, nullptr, 1.0f, Dq, N8, h);
  k_geglu<<<(Bq * N4) / 256, 256, 0, stream>>>(h, act);
  // xfin = xnew + 0.1*(act @ W2 + b2)   (64 N-tiles -> 8 blocks)
  k_gemv_wmma<<<(Dq / 16) / 8, 256, 0, stream>>>(act, W2, b2, xnew, 0.1f, N4, Dq, xfin);
  k_logits<<<(Vq + 7) / 8, 256, 0, stream>>>(xfin, te, out);
}